// CritiGraph_47433618817071
// MI455X (gfx1250) — compile-verified
//
#include <hip/hip_runtime.h>

// ---------------- problem constants (from reference) ----------------
#define H_   16
#define K_   64
#define HK   1024          // H*K
#define M_   2049          // 2*H*K + 1
#define TP_  8
#define T1_  16
#define T2_  32
#define B_   4
#define NT   64            // B*T1
#define T2_PER_BLK 4
#define THREADS 256
#define RAWN (HK * TP_)    // 8192 rand dwords per (b,t1)

typedef __attribute__((ext_vector_type(4))) float    vf4;
typedef __attribute__((ext_vector_type(4))) unsigned vu4;

// Async global->LDS staging (ASYNCcnt path) if this toolchain exposes it.
#if defined(__HIP_DEVICE_COMPILE__) && defined(__gfx1250__)
#if __has_builtin(__builtin_amdgcn_global_load_async_to_lds_b32)
#define USE_ASYNC 1
#endif
#endif

#ifdef USE_ASYNC
typedef __attribute__((address_space(1))) int g1_int;
typedef __attribute__((address_space(3))) int l3_int;
#endif

__global__ __launch_bounds__(THREADS) void critigraph_kernel(
    const int*   __restrict__ sta,    // [NT][TP]
    const int*   __restrict__ pos,    // [B][T2][TP]
    const float* __restrict__ valn,   // [NT][T2]  ([B][T1][T2])
    const int*   __restrict__ randv,  // [NT][H][K][TP]
    const int*   __restrict__ perm,   // [M]
    float*       __restrict__ out)    // [NT][T2][M][TP]
{
    __shared__ vu4            s_resv[HK + 1];             // [HK+1][TP] u16 mags; row HK = |sta|
    __shared__ unsigned short s_perm[M_ + 7];
    __shared__ float          s_dsta[T2_PER_BLK * TP_];
    __shared__ float          s_base[T2_PER_BLK * TP_];   // dis_sum - dis_sta
    __shared__ float          s_coef[T2_PER_BLK * TP_];   // sign(pos)*val_n
    __shared__ unsigned       s_pabs[T2_PER_BLK * TP_];
#ifdef USE_ASYNC
    __shared__ unsigned       s_raw[RAWN];                // raw rand dwords, 32 KB
#endif

    const int tid    = threadIdx.x;
    const int i      = blockIdx.x;          // b*T1 + t1
    const int b      = i >> 4;
    const int t2base = blockIdx.y * T2_PER_BLK;
    const int rbase  = i * RAWN;            // 8192 ints per (b,t1)

    // gfx1250 prefetch (global_prefetch_b8) to warm L2 for this block's slice
    __builtin_prefetch(randv + rbase + tid * 32, 0, 0);
    __builtin_prefetch(perm + (tid & 255), 0, 0);

#ifdef USE_ASYNC
    // Stage the 8192 rand dwords for this (b,t1) into LDS asynchronously.
    // Constant trip count -> fully unrolled straight-line issues (no exec-mask
    // loop skeleton); 32 outstanding ASYNCcnt ops per wave (limit is 63).
#pragma unroll
    for (int r = 0; r < RAWN / THREADS; ++r) {
        const int q = tid + r * THREADS;
        __builtin_amdgcn_global_load_async_to_lds_b32(
            (g1_int*)(int*)(randv + rbase + q),
            (l3_int*)(int*)(&s_raw[q]),
            0, 0);
    }
#endif

    // |sta| and sign(sta) per tree, in registers (overlaps async staging)
    unsigned oriv[TP_];
    float    ssgn[TP_];
#pragma unroll
    for (int t = 0; t < TP_; ++t) {
        int sv  = sta[i * TP_ + t];
        oriv[t] = (unsigned)(sv < 0 ? -sv : sv);
        ssgn[t] = (sv >= 0) ? 1.0f : -1.0f;
    }

    // perm table -> LDS (u16)
    for (int m = tid; m < M_; m += THREADS)
        s_perm[m] = (unsigned short)perm[m];

    // row HK of the candidate table = |sta| (removes the src==1024 branch)
    if (tid < TP_)
        ((unsigned short*)s_resv)[HK * TP_ + tid] = (unsigned short)oriv[tid];

    // per-(t2,t) constants: |pos|, sign(pos)*val_n, dis_sta
    if (tid < T2_PER_BLK * TP_) {
        const int t2l = tid >> 3, t = tid & 7;
        const int t2  = t2base + t2l;
        const int pv  = pos[(b * T2_ + t2) * TP_ + t];
        const unsigned pa = (unsigned)(pv < 0 ? -pv : pv);
        const float psg   = (pv >= 0) ? 1.0f : -1.0f;
        const float norm  = valn[i * T2_ + t2];
        const unsigned x  = oriv[t] ^ pa;
        const int bl      = 32 - __builtin_clz(x + 1u);   // floor(log2(x+1))+1
        const float sg    = ssgn[t] * psg;
        s_dsta[tid] = sg * (1.0f - (float)bl * 0.0625f) * norm;
        s_pabs[tid] = pa;
        s_coef[tid] = psg * norm;
    }
    __syncthreads();
    if (tid < T2_PER_BLK * TP_) {
        const int t2l = tid >> 3;
        float dsum = 0.0f;
#pragma unroll
        for (int tt = 0; tt < TP_; ++tt) dsum += s_dsta[t2l * TP_ + tt];
        s_base[tid] = dsum - s_dsta[tid];
    }

    // Candidate magnitude table: res[j*K+k][t] = (|sta_t| ^ (1<<j)) ^ (rand & ((1<<j)-1))
    unsigned short* s_res = (unsigned short*)s_resv;
#ifdef USE_ASYNC
#if __has_builtin(__builtin_amdgcn_s_wait_asynccnt)
    __builtin_amdgcn_s_wait_asynccnt(0);
#else
    asm volatile("s_wait_asynccnt 0" ::: "memory");
#endif
#pragma unroll 8
    for (int r = 0; r < RAWN / THREADS; ++r) {
        const int q = tid + r * THREADS;
        const unsigned rr  = s_raw[q];
        const int      j   = q >> 9;                 // q = ((j*64)+k)*8 + t
        const unsigned msk = rr & ((1u << j) - 1u);
        s_res[q] = (unsigned short)((oriv[q & 7] ^ (1u << j)) ^ msk);
    }
#else
#pragma unroll 8
    for (int r = 0; r < RAWN / THREADS; ++r) {
        const int q = tid + r * THREADS;
        const unsigned rr  = (unsigned)randv[rbase + q];
        const int      j   = q >> 9;
        const unsigned msk = rr & ((1u << j) - 1u);
        s_res[q] = (unsigned short)((oriv[q & 7] ^ (1u << j)) ^ msk);
    }
#endif
    __syncthreads();

    // ---------------- main streaming loop ----------------
    for (int t2l = 0; t2l < T2_PER_BLK; ++t2l) {
        const int t2 = t2base + t2l;
        float    baseR[TP_], coefR[TP_];
        unsigned paR[TP_];
#pragma unroll
        for (int t = 0; t < TP_; ++t) {
            baseR[t] = s_base[t2l * TP_ + t];
            coefR[t] = s_coef[t2l * TP_ + t];
            paR[t]   = s_pabs[t2l * TP_ + t];
        }
        vf4* op = (vf4*)(out + (size_t)(i * T2_ + t2) * (size_t)(M_ * TP_));

        for (int m = tid; m < M_; m += THREADS) {
            const int src  = s_perm[m];
            const bool neg = src > HK;                 // '-result' half
            const int ridx = neg ? (src - (HK + 1)) : src;   // src==HK -> |sta| row
            const vu4 rv = s_resv[ridx];               // ds_load_b128: 8 packed u16
            unsigned mg[TP_];
            mg[0] = rv.x & 0xFFFFu; mg[1] = rv.x >> 16;
            mg[2] = rv.y & 0xFFFFu; mg[3] = rv.y >> 16;
            mg[4] = rv.z & 0xFFFFu; mg[5] = rv.z >> 16;
            mg[6] = rv.w & 0xFFFFu; mg[7] = rv.w >> 16;

            float o[TP_];
#pragma unroll
            for (int t = 0; t < TP_; ++t) {
                float c = coefR[t];
                if (neg && mg[t] != 0u) c = -c;   // sign(cnc): -res<0 unless res==0
                const unsigned x = mg[t] ^ paR[t];
                const int bl = 32 - __builtin_clz(x + 1u);
                const float val = c * (1.0f - (float)bl * 0.0625f);
                o[t] = (baseR[t] + val) * 0.125f; // (dsum - dsta + dcnc)/TP
            }
            const vf4 o0 = {o[0], o[1], o[2], o[3]};
            const vf4 o1 = {o[4], o[5], o[6], o[7]};
            __builtin_nontemporal_store(o0, op + (size_t)m * 2);
            __builtin_nontemporal_store(o1, op + (size_t)m * 2 + 1);
        }
    }
}

extern "C" void kernel_launch(void* const* d_in, const int* in_sizes, int n_in,
                              void* d_out, int out_size, void* d_ws, size_t ws_size,
                              hipStream_t stream) {
    const int*   sta   = (const int*)d_in[0];
    const int*   pos   = (const int*)d_in[1];
    const float* valn  = (const float*)d_in[2];
    const int*   randv = (const int*)d_in[3];
    const int*   perm  = (const int*)d_in[4];
    float*       out   = (float*)d_out;

    dim3 grid(NT, T2_ / T2_PER_BLK);   // 64 x 8 = 512 workgroups
    critigraph_kernel<<<grid, dim3(THREADS), 0, stream>>>(
        sta, pos, valn, randv, perm, out);
}